// DepGraph_10230612099246
// MI455X (gfx1250) — compile-verified
//
#include <hip/hip_runtime.h>
#include <hip/hip_bf16.h>
#include <math.h>

typedef __attribute__((ext_vector_type(2))) float v2f;
typedef __attribute__((ext_vector_type(8))) float v8f;

#define MDIM 16384
#define NDIM 4096
#define KD   64
#define BM   128
#define BN   128
#define LDA  68   // padded LDS stride in floats: 68*4=272 bytes (16B aligned, conflict-free)

// logitexp(a) for a<=0 collapses to a - log(1 - e^a): the reference's pos/neg
// split evaluates this same function piecewise (neg_val==0 when a>=-log2,
// pos_val==0 when a<-log2). Operand range here is a ~ -8, so 1-e^a is well
// conditioned. Use hardware v_exp_f32/v_log_f32 (TRANS, co-executes with VALU).
__device__ __forceinline__ float pair_value(float aexp, float u) {
    float ea      = __expf(aexp);
    float logits  = aexp - __logf(fmaxf(1.0f - ea, 1e-20f));
    float logistic = __logf(u) - __logf(1.0f - u);
    float x = (logits + logistic) * (1.0f / 0.3f);
    return __builtin_amdgcn_rcpf(1.0f + __expf(-x));   // sigmoid via v_rcp_f32
}

// ---- ordering[p] = sum_d log(0.5 + 0.5*erf(uR[p,d]/sqrt(2))) : one wave32 per row
__global__ __launch_bounds__(256) void ordering_kernel(const float* __restrict__ uR,
                                                       float* __restrict__ ordering) {
    int wave = (blockIdx.x * blockDim.x + threadIdx.x) >> 5;  // row id, 0..4095
    int lane = threadIdx.x & 31;
    const float INV_SQRT2 = 0.70710678118654752440f;
    float x0 = uR[wave * KD + lane];
    float x1 = uR[wave * KD + 32 + lane];
    float s = __logf(0.5f + 0.5f * erff(x0 * INV_SQRT2)) +
              __logf(0.5f + 0.5f * erff(x1 * INV_SQRT2));
    #pragma unroll
    for (int off = 16; off >= 1; off >>= 1) s += __shfl_xor(s, off, 32);
    if (lane == 0) ordering[wave] = s;
}

// ---- rank[p] = stable argsort position of p (== inverse permutation)
__global__ __launch_bounds__(256) void rank_kernel(const float* __restrict__ ordering,
                                                   int* __restrict__ rank) {
    __shared__ float sOrd[NDIM];
    for (int i = threadIdx.x; i < NDIM; i += blockDim.x) sOrd[i] = ordering[i];
    __syncthreads();
    int p = blockIdx.x * blockDim.x + threadIdx.x;
    float op = sOrd[p];
    int r = 0;
    for (int q = 0; q < NDIM; ++q) {
        float oq = sOrd[q];
        r += (oq < op) || (oq == op && q < p);
    }
    rank[p] = r;
}

// ---- 128x128-tile FP32 WMMA gram kernel + relaxed-Bernoulli epilogue.
// GMODE=false: out[m,n] = rb(logits(uM_m, uR_n), noise[m,n])
// GMODE=true : out[p,q] = rank[p]<rank[q] ? rb(logits(uR_p,uR_q), noise[rank[p],rank[q]]) : 0
template <bool GMODE>
__global__ __launch_bounds__(256) void pair_kernel(const float* __restrict__ Lmat,
                                                   const float* __restrict__ Rmat,
                                                   const float* __restrict__ g_logscale,
                                                   const float* __restrict__ noise,
                                                   const int* __restrict__ rank,
                                                   float* __restrict__ out) {
    __shared__ float As[BM * LDA];
    __shared__ float Bs[BN * LDA];
    __shared__ float nA[BM];
    __shared__ float nB[BN];
    __shared__ int   rP[BM];
    __shared__ int   rQ[BN];

    const int t = threadIdx.x;
    const int m0 = blockIdx.y * BM;
    const int n0 = blockIdx.x * BN;

    // ---- A-mode: prefetch the coalesced 64KB noise tile while we do WMMA ----
    if (!GMODE) {
        #pragma unroll
        for (int i = 0; i < 2; ++i) {
            int line = i * 256 + t;            // 512 x 128B lines per tile
            int row  = line >> 2;
            int col  = (line & 3) << 5;        // 32 floats per 128B line
            __builtin_prefetch(noise + (size_t)(m0 + row) * NDIM + n0 + col, 0, 1);
        }
    }

    // ---- stage 128x64 slabs of both operands into LDS (coalesced float4) ----
    #pragma unroll
    for (int i = 0; i < 8; ++i) {
        int idx = i * 256 + t;          // 0..2047 float4 slots per slab
        int row = idx >> 4;             // 16 float4 per row
        int c4  = (idx & 15) << 2;      // float column
        float4 va = *(const float4*)(Lmat + (size_t)(m0 + row) * KD + c4);
        *(float4*)&As[row * LDA + c4] = va;
        float4 vb = *(const float4*)(Rmat + (size_t)(n0 + row) * KD + c4);
        *(float4*)&Bs[row * LDA + c4] = vb;
    }
    __syncthreads();

    // ---- per-row squared norms from LDS (gram trick) + rank staging ----
    {
        int row = t & 127;
        const float* src = (t < 128) ? &As[row * LDA] : &Bs[row * LDA];
        float s = 0.0f;
        #pragma unroll
        for (int k = 0; k < KD; ++k) { float v = src[k]; s = fmaf(v, v, s); }
        if (t < 128) nA[row] = s; else nB[row] = s;
        if (GMODE) {
            if (t < 128) rP[row] = rank[m0 + row];
            else         rQ[row] = rank[n0 + row];
        }
    }
    __syncthreads();

    // ---- WMMA main loop: 8 waves in a 4(M) x 2(N) grid, each owns 32x64 ----
    const int lane  = t & 31;
    const int wave  = t >> 5;
    const int waveM = wave >> 1;      // 0..3
    const int waveN = wave & 1;       // 0..1
    const int half  = lane >> 4;      // 0 | 1
    const int l15   = lane & 15;
    const int mBase = waveM * 32;
    const int nBase = waveN * 64;

    v8f acc[2][4] = {};
    for (int k0 = 0; k0 < KD; k0 += 4) {
        int kk = k0 + 2 * half;       // VGPR0 holds K=kk, VGPR1 holds K=kk+1 (ISA layout)
        v2f a[2], b[4];
        #pragma unroll
        for (int mt = 0; mt < 2; ++mt)
            a[mt] = *(const v2f*)&As[(mBase + mt * 16 + l15) * LDA + kk];
        #pragma unroll
        for (int nt = 0; nt < 4; ++nt)
            b[nt] = *(const v2f*)&Bs[(nBase + nt * 16 + l15) * LDA + kk];
        #pragma unroll
        for (int mt = 0; mt < 2; ++mt)
            #pragma unroll
            for (int nt = 0; nt < 4; ++nt)
                acc[mt][nt] = __builtin_amdgcn_wmma_f32_16x16x4_f32(
                    false, a[mt], false, b[nt], (short)0, acc[mt][nt], false, false);
    }

    // ---- epilogue ----
    const float inv_g = __expf(-g_logscale[0]);
    #pragma unroll
    for (int mt = 0; mt < 2; ++mt) {
        #pragma unroll
        for (int nt = 0; nt < 4; ++nt) {
            #pragma unroll
            for (int r = 0; r < 8; ++r) {
                int ml = mBase + mt * 16 + r + 8 * half;  // C/D layout: M = r + 8*half
                int nl = nBase + nt * 16 + l15;           // N = lane & 15
                int gm = m0 + ml;
                int gn = n0 + nl;
                float dot = acc[mt][nt][r];
                float sq  = nA[ml] + nB[nl] - 2.0f * dot;
                float ae  = -0.5f * sq * inv_g;
                float val;
                if (GMODE) {
                    int rp = rP[ml], rq = rQ[nl];
                    val = 0.0f;
                    if (rp < rq) {
                        float u = noise[(size_t)rp * NDIM + rq];
                        val = pair_value(ae, u);
                    }
                } else {
                    float u = noise[(size_t)gm * NDIM + gn];
                    val = pair_value(ae, u);
                }
                out[(size_t)gm * NDIM + gn] = val;
            }
        }
    }
}

extern "C" void kernel_launch(void* const* d_in, const int* in_sizes, int n_in,
                              void* d_out, int out_size, void* d_ws, size_t ws_size,
                              hipStream_t stream) {
    (void)in_sizes; (void)n_in; (void)out_size; (void)ws_size;
    const float* uM      = (const float*)d_in[0];
    const float* uR      = (const float*)d_in[1];
    const float* gls     = (const float*)d_in[2];
    const float* noise_A = (const float*)d_in[3];
    const float* noise_G = (const float*)d_in[4];
    float* outA = (float*)d_out;
    float* outG = outA + (size_t)MDIM * NDIM;

    float* ordering = (float*)d_ws;
    int*   rank     = (int*)((char*)d_ws + NDIM * sizeof(float));

    // ordering: 4096 rows, one wave each -> 512 blocks of 256 (8 waves)
    ordering_kernel<<<NDIM / 8, 256, 0, stream>>>(uR, ordering);
    // stable ranks (== inverse permutation of argsort)
    rank_kernel<<<NDIM / 256, 256, 0, stream>>>(ordering, rank);

    // A: 16384 x 4096
    dim3 gridA(NDIM / BN, MDIM / BM);
    pair_kernel<false><<<gridA, 256, 0, stream>>>(uM, uR, gls, noise_A, nullptr, outA);
    // G: 4096 x 4096
    dim3 gridG(NDIM / BN, NDIM / BM);
    pair_kernel<true><<<gridG, 256, 0, stream>>>(uR, uR, gls, noise_G, rank, outG);
}